// ContrastiveLoss_3075196584402
// MI455X (gfx1250) — compile-verified
//
#include <hip/hip_runtime.h>

// ContrastiveLoss (symmetric InfoNCE) fused for MI455X / gfx1250.
// N=8192, D=256 fp32. Workspace use: 2*N*D floats (Z1n,Z2n) + 7*N floats
// of reduction arrays  ~= 16.2 MB  (well under typical ws_size).

#define D_DIM   256
#define TAU_INV 2.0f      // 1/tau, tau = 0.5

typedef __attribute__((ext_vector_type(2))) float v2f;
typedef __attribute__((ext_vector_type(4))) float v4f;
typedef __attribute__((ext_vector_type(8))) float v8f;

// D = A(16x4,f32) * B(4x16,f32) + C(16x16,f32)  -- native fp32 WMMA on CDNA5.
// 8-arg pattern per probe doc: (neg_a, A, neg_b, B, c_mod, C, reuse_a, reuse_b)
__device__ __forceinline__ v8f wmma_f32x4(v2f a, v2f b, v8f c) {
  return __builtin_amdgcn_wmma_f32_16x16x4_f32(false, a, false, b, (short)0, c,
                                               false, false);
}

// ---------------------------------------------------------------------------
// Stage 0: zero the atomic-accumulation arrays (rowsum11/22/12, colsum12).
// ---------------------------------------------------------------------------
__global__ void zero_kernel(float* __restrict__ p, int n) {
  int i = blockIdx.x * blockDim.x + threadIdx.x;
  if (i < n) p[i] = 0.0f;
}

// ---------------------------------------------------------------------------
// Stage 1: row L2 normalization, one wave (32 lanes) per row, 8 floats/lane.
// z = x / max(||x||, 1e-12)   (torch F.normalize semantics)
// ---------------------------------------------------------------------------
__global__ __launch_bounds__(256) void normalize_rows(
    const float* __restrict__ X1, const float* __restrict__ X2,
    float* __restrict__ Z1, float* __restrict__ Z2, int N) {
  int gw   = (blockIdx.x * blockDim.x + threadIdx.x) >> 5;  // global wave id
  int lane = threadIdx.x & 31;
  if (gw >= 2 * N) return;
  const float* X = (gw < N) ? X1 : X2;
  float*       Z = (gw < N) ? Z1 : Z2;
  int row = (gw < N) ? gw : gw - N;

  const v4f* src = (const v4f*)(X + (size_t)row * D_DIM);
  v4f*       dst = (v4f*)(Z + (size_t)row * D_DIM);
  v4f u = src[lane];        // floats [4*lane .. 4*lane+3]
  v4f w = src[lane + 32];   // floats [128+4*lane .. ]

  float ss = u[0]*u[0] + u[1]*u[1] + u[2]*u[2] + u[3]*u[3]
           + w[0]*w[0] + w[1]*w[1] + w[2]*w[2] + w[3]*w[3];
  #pragma unroll
  for (int m = 1; m < 32; m <<= 1) ss += __shfl_xor(ss, m, 32);

  float inv = 1.0f / fmaxf(sqrtf(ss), 1e-12f);
  dst[lane]      = u * inv;
  dst[lane + 32] = w * inv;
}

// ---------------------------------------------------------------------------
// Stage 2: fused  S = ZA * ZB^T ;  E = exp(S/tau) ; row sums (+ optional
// column sums) of E ; diagonal capture.  Each wave -> 32x32 tile (2x2 WMMA
// frags), 4-wave block -> 64x64 tile.  K = 256 in steps of 4.
//
// Fragment layouts (ISA 7.12.2, f32):
//   A 16x4 : lane m (m=lane&15) holds row m; VGPR0/1 = K {0,1} (lanes 0-15)
//            or K {2,3} (lanes 16-31)  -> one float2 load per lane.
//   B 4x16 : lane n holds col n with the same K striping -> same load shape.
//   C 16x16: VGPR v, lanes 0-15 -> (M=v, N=lane); lanes 16-31 -> (M=v+8).
// ---------------------------------------------------------------------------
__global__ __launch_bounds__(128) void gram_exp_sums(
    const float* __restrict__ ZA, const float* __restrict__ ZB,
    float* __restrict__ rowsum, float* __restrict__ colsum,
    float* __restrict__ diag_e, float* __restrict__ diag_s) {
  const int lane = threadIdx.x & 31;
  const int wave = threadIdx.x >> 5;
  const int l    = lane & 15;
  const int half = lane >> 4;

  const int rb = blockIdx.y * 64 + (wave >> 1) * 32;
  const int cb = blockIdx.x * 64 + (wave & 1) * 32;

  const float* Arow0 = ZA + (size_t)(rb + l) * D_DIM + 2 * half;
  const float* Arow1 = Arow0 + (size_t)16 * D_DIM;
  const float* Brow0 = ZB + (size_t)(cb + l) * D_DIM + 2 * half;
  const float* Brow1 = Brow0 + (size_t)16 * D_DIM;

  v8f acc00 = {}, acc01 = {}, acc10 = {}, acc11 = {};

  #pragma unroll 4
  for (int k0 = 0; k0 < D_DIM; k0 += 4) {
    v2f a0 = *(const v2f*)(Arow0 + k0);
    v2f a1 = *(const v2f*)(Arow1 + k0);
    v2f b0 = *(const v2f*)(Brow0 + k0);
    v2f b1 = *(const v2f*)(Brow1 + k0);
    acc00 = wmma_f32x4(a0, b0, acc00);
    acc01 = wmma_f32x4(a0, b1, acc01);
    acc10 = wmma_f32x4(a1, b0, acc10);
    acc11 = wmma_f32x4(a1, b1, acc11);
  }

  // ---- epilogue: exp(S/tau), diagonal capture (each diag element owned by
  // exactly one wave -> plain stores are race-free).
  auto apply_exp = [&](v8f& acc, int r0, int c0) {
    #pragma unroll
    for (int v = 0; v < 8; ++v) {
      float s = acc[v] * TAU_INV;
      float e = __expf(s);
      int grow = r0 + v + 8 * half;
      int gcol = c0 + l;
      if (grow == gcol) {
        if (diag_e) diag_e[grow] = e;
        if (diag_s) diag_s[grow] = s;
      }
      acc[v] = e;
    }
  };
  apply_exp(acc00, rb,      cb);
  apply_exp(acc01, rb,      cb + 16);
  apply_exp(acc10, rb + 16, cb);
  apply_exp(acc11, rb + 16, cb + 16);

  // ---- row sums: rows live in 16-lane halves; butterfly-reduce width 16.
  auto row_reduce = [&](v8f e0, v8f e1, int r0) {
    #pragma unroll
    for (int v = 0; v < 8; ++v) {
      float t = e0[v] + e1[v];
      t += __shfl_xor(t, 1, 16);
      t += __shfl_xor(t, 2, 16);
      t += __shfl_xor(t, 4, 16);
      t += __shfl_xor(t, 8, 16);
      if (l == 0) atomicAdd(&rowsum[r0 + v + 8 * half], t);
    }
  };
  row_reduce(acc00, acc01, rb);
  row_reduce(acc10, acc11, rb + 16);

  // ---- column sums (S12 only): fixed column per lane; combine halves.
  if (colsum) {
    auto col_reduce = [&](v8f e0, v8f e1, int c0) {
      float t = 0.0f;
      #pragma unroll
      for (int v = 0; v < 8; ++v) t += e0[v] + e1[v];
      t += __shfl_xor(t, 16, 32);          // lane L and L+16 share a column
      if (half == 0) atomicAdd(&colsum[c0 + l], t);
    };
    col_reduce(acc00, acc10, cb);
    col_reduce(acc01, acc11, cb + 16);
  }
}

// ---------------------------------------------------------------------------
// Stage 3: loss = (sum_i log den1_i + log den2_i - 2*s12_ii) / (2N)
//   den1_i = rowsum(exp S11)_i - exp(S11_ii/tau) + rowsum(exp S12)_i
//   den2_i = rowsum(exp S22)_i - exp(S22_ii/tau) + colsum(exp S12)_i
// ---------------------------------------------------------------------------
__global__ __launch_bounds__(256) void finalize_loss(
    const float* __restrict__ row11, const float* __restrict__ row22,
    const float* __restrict__ row12, const float* __restrict__ col12,
    const float* __restrict__ e11, const float* __restrict__ e22,
    const float* __restrict__ s12, float* __restrict__ out, int N) {
  double acc = 0.0;
  for (int i = threadIdx.x; i < N; i += 256) {
    float den1 = row11[i] - e11[i] + row12[i];
    float den2 = row22[i] - e22[i] + col12[i];
    acc += (double)logf(den1) + (double)logf(den2) - 2.0 * (double)s12[i];
  }
  __shared__ double sm[256];
  sm[threadIdx.x] = acc;
  __syncthreads();
  for (int s = 128; s > 0; s >>= 1) {
    if (threadIdx.x < s) sm[threadIdx.x] += sm[threadIdx.x + s];
    __syncthreads();
  }
  if (threadIdx.x == 0) out[0] = (float)(sm[0] / (2.0 * (double)N));
}

// ---------------------------------------------------------------------------
extern "C" void kernel_launch(void* const* d_in, const int* in_sizes, int n_in,
                              void* d_out, int out_size, void* d_ws,
                              size_t ws_size, hipStream_t stream) {
  const float* H1 = (const float*)d_in[0];
  const float* H2 = (const float*)d_in[1];
  float* out = (float*)d_out;

  const int N = in_sizes[0] / D_DIM;   // 8192

  float* ws    = (float*)d_ws;
  float* Z1    = ws;                       // N*D
  float* Z2    = Z1 + (size_t)N * D_DIM;   // N*D
  float* row11 = Z2 + (size_t)N * D_DIM;   // N
  float* row22 = row11 + N;                // N
  float* row12 = row22 + N;                // N
  float* col12 = row12 + N;                // N (contiguous with rows -> one memzero)
  float* e11   = col12 + N;                // N (fully overwritten, no zeroing)
  float* e22   = e11 + N;                  // N
  float* s12   = e22 + N;                  // N

  // Stage 0: zero atomic accumulators (4*N floats, contiguous).
  zero_kernel<<<(4 * N + 255) / 256, 256, 0, stream>>>(row11, 4 * N);

  // Stage 1: normalize rows of both inputs (2N waves, 8 waves/block).
  {
    int waves  = 2 * N;
    int blocks = (waves * 32 + 255) / 256;
    normalize_rows<<<blocks, 256, 0, stream>>>(H1, H2, Z1, Z2, N);
  }

  // Stage 2: three fused Gram-exp-reduce passes.
  dim3 grid(N / 64, N / 64);
  gram_exp_sums<<<grid, 128, 0, stream>>>(Z1, Z1, row11, nullptr, e11, nullptr);
  gram_exp_sums<<<grid, 128, 0, stream>>>(Z2, Z2, row22, nullptr, e22, nullptr);
  gram_exp_sums<<<grid, 128, 0, stream>>>(Z1, Z2, row12, col12, nullptr, s12);

  // Stage 3: scalar loss.
  finalize_loss<<<1, 256, 0, stream>>>(row11, row22, row12, col12, e11, e22,
                                       s12, out, N);
}